// ModifiedGCNFN_9990093930997
// MI455X (gfx1250) — compile-verified
//
#include <hip/hip_runtime.h>
#include <math.h>

#define N_NODES 100000
#define N_EDGES 3200000
#define IN_CH   128
#define HID     64
#define N_GRAPHS 64
#define NEG_SLOPE 0.2f

#define SELU_SCALE 1.0507009873554805f
#define SELU_ALPHA 1.6732632423543772f

typedef __attribute__((ext_vector_type(2))) float v2f;
typedef __attribute__((ext_vector_type(8))) float v8f;

// ---------------- helpers ----------------

__device__ __forceinline__ void atomicMaxF32(float* addr, float val) {
    // monotonic mapping trick; storage initialized to -inf (0xFF800000)
    int iv = __float_as_int(val);
    if (iv >= 0) {
        atomicMax((int*)addr, iv);
    } else {
        atomicMin((unsigned int*)addr, (unsigned int)iv);
    }
}

__device__ __forceinline__ float seluf(float v) {
    return SELU_SCALE * (v > 0.0f ? v : SELU_ALPHA * (__expf(v) - 1.0f));
}

// ---------------- fill ----------------

__global__ void fill_f32(float* __restrict__ p, float v, int n) {
    int i = blockIdx.x * blockDim.x + threadIdx.x;
    if (i < n) p[i] = v;
}

// ---------------- WMMA GEMM: C[M,64] = A[M,K] @ B[K,64] (fp32 exact) ----------------
// One wave computes one 16x16 tile of C using V_WMMA_F32_16X16X4_F32.
// A fragment (16x4 f32): lanes 0-15 hold row M=lane, K = 0,1; lanes 16-31 hold K = 2,3.
// B fragment (4x16 f32): mirrored, lane -> column N, vgpr0/1 -> K pair per half.
// C/D (16x16 f32): VGPR i -> row i (lanes 0-15) / i+8 (lanes 16-31), N = lane&15.
template <int K>
__global__ void gemm_wmma_f32(const float* __restrict__ A,
                              const float* __restrict__ B,
                              float* __restrict__ C, int M) {
    int wave  = (blockIdx.x * blockDim.x + threadIdx.x) >> 5;
    int lane  = threadIdx.x & 31;
    int tilesM = M >> 4;                 // M multiple of 16
    int tileN = wave & 3;                // 64 / 16 = 4 tiles in N
    int tileM = wave >> 2;
    if (tileM >= tilesM) return;         // wave-uniform: EXEC stays all-ones inside

    int half = lane >> 4;
    int r    = lane & 15;

    const float* Arow = A + (size_t)(tileM * 16 + r) * K;
    const float* Bcol = B + (tileN * 16 + r);

    v8f acc = {};
#pragma unroll 4
    for (int k = 0; k < K; k += 4) {
        int kk = k + 2 * half;
        v2f a, b;
        a.x = Arow[kk];
        a.y = Arow[kk + 1];
        b.x = Bcol[kk * HID];
        b.y = Bcol[(kk + 1) * HID];
        acc = __builtin_amdgcn_wmma_f32_16x16x4_f32(
            /*neg_a=*/false, a, /*neg_b=*/false, b,
            /*c_mod=*/(short)0, acc, /*reuse_a=*/false, /*reuse_b=*/false);
    }

    float* Crow = C + (size_t)(tileM * 16) * HID + tileN * 16 + r;
#pragma unroll
    for (int i = 0; i < 8; ++i) {
        Crow[(size_t)(i + 8 * half) * HID] = acc[i];
    }
}

// ---------------- per-node dots: el = h@a1, er = h@a2 (wave per node) ----------------

__global__ void node_dots(const float* __restrict__ h,
                          const float* __restrict__ a1,
                          const float* __restrict__ a2,
                          float* __restrict__ o1, float* __restrict__ o2, int n) {
    int wave = (blockIdx.x * blockDim.x + threadIdx.x) >> 5;
    int lane = threadIdx.x & 31;
    if (wave >= n) return;
    const float* row = h + (size_t)wave * HID;
    float x0 = row[lane], x1 = row[lane + 32];
    float p1 = x0 * a1[lane] + x1 * a1[lane + 32];
    float p2 = x0 * a2[lane] + x1 * a2[lane + 32];
    for (int off = 16; off > 0; off >>= 1) {
        p1 += __shfl_down(p1, off, 32);
        p2 += __shfl_down(p2, off, 32);
    }
    if (lane == 0) { o1[wave] = p1; o2[wave] = p2; }
}

// ---------------- edge passes ----------------

__global__ void edge_max(const int* __restrict__ src, const int* __restrict__ dst,
                         const float* __restrict__ el, const float* __restrict__ er,
                         float* __restrict__ m, int nE) {
    int e = blockIdx.x * blockDim.x + threadIdx.x;
    if (e >= nE) return;
    int s = src[e], d = dst[e];
    float v = el[s] + er[d];
    v = v > 0.0f ? v : NEG_SLOPE * v;
    atomicMaxF32(&m[d], v);
}

__global__ void edge_exp(const int* __restrict__ src, const int* __restrict__ dst,
                         const float* __restrict__ el, const float* __restrict__ er,
                         const float* __restrict__ m, float* __restrict__ exb,
                         float* __restrict__ sbuf, int nE) {
    int e = blockIdx.x * blockDim.x + threadIdx.x;
    if (e >= nE) return;
    int s = src[e], d = dst[e];
    float v = el[s] + er[d];
    v = v > 0.0f ? v : NEG_SLOPE * v;
    float ex = __expf(v - m[d]);
    exb[e] = ex;
    atomicAdd(&sbuf[d], ex);
}

// one thread per (edge, channel): coalesced gather of h[src] and atomic scatter to out[dst]
__global__ void edge_agg(const int* __restrict__ src, const int* __restrict__ dst,
                         const float* __restrict__ exb, const float* __restrict__ sbuf,
                         const float* __restrict__ h, float* __restrict__ out, int nE) {
    int idx = blockIdx.x * blockDim.x + threadIdx.x;
    int e = idx >> 6;
    int c = idx & 63;
    if (e >= nE) return;
    int s = src[e], d = dst[e];
    float a = exb[e] / sbuf[d];
    atomicAdd(&out[(size_t)d * HID + c], a * h[(size_t)s * HID + c]);
}

// ---------------- bias + SELU epilogue (in place) ----------------

__global__ void bias_selu(float* __restrict__ h, const float* __restrict__ b, int total) {
    int i = blockIdx.x * blockDim.x + threadIdx.x;
    if (i >= total) return;
    int c = i & 63;
    h[i] = seluf(h[i] + b[c]);
}

// ---------------- pooling ----------------

__global__ void gate_compute(const float* __restrict__ h, const float* __restrict__ gw,
                             const float* __restrict__ gb, const int* __restrict__ ng,
                             float* __restrict__ gate, float* __restrict__ gm, int n) {
    int wave = (blockIdx.x * blockDim.x + threadIdx.x) >> 5;
    int lane = threadIdx.x & 31;
    if (wave >= n) return;
    const float* row = h + (size_t)wave * HID;
    float p = row[lane] * gw[lane] + row[lane + 32] * gw[lane + 32];
    for (int off = 16; off > 0; off >>= 1) p += __shfl_down(p, off, 32);
    if (lane == 0) {
        float g = p + gb[0];
        gate[wave] = g;
        atomicMaxF32(&gm[ng[wave]], g);
    }
}

__global__ void gate_exp(float* __restrict__ gate, const float* __restrict__ gm,
                         float* __restrict__ gs, const int* __restrict__ ng, int n) {
    int i = blockIdx.x * blockDim.x + threadIdx.x;
    if (i >= n) return;
    int g = ng[i];
    float ge = __expf(gate[i] - gm[g]);
    gate[i] = ge;
    atomicAdd(&gs[g], ge);
}

__global__ void pool_agg(const float* __restrict__ gate, const float* __restrict__ gs,
                         const float* __restrict__ h, const int* __restrict__ ng,
                         float* __restrict__ pooled, int n) {
    int idx = blockIdx.x * blockDim.x + threadIdx.x;
    int i = idx >> 6;
    int c = idx & 63;
    if (i >= n) return;
    int g = ng[i];
    float alpha = gate[i] / gs[g];
    atomicAdd(&pooled[(size_t)g * HID + c], alpha * h[(size_t)i * HID + c]);
}

// ---------------- MLP head: out = sigmoid(selu(pooled@fc1+b1)@fc2+b2) ----------------

__global__ void head_kernel(const float* __restrict__ pooled,
                            const float* __restrict__ fc1W, const float* __restrict__ fc1b,
                            const float* __restrict__ fc2W, const float* __restrict__ fc2b,
                            float* __restrict__ out) {
    int g = threadIdx.x;
    if (g >= N_GRAPHS) return;
    const float* p = pooled + (size_t)g * HID;
    float z[HID / 2];
#pragma unroll
    for (int j = 0; j < HID / 2; ++j) {
        float acc = fc1b[j];
        for (int c = 0; c < HID; ++c) acc += p[c] * fc1W[c * (HID / 2) + j];
        z[j] = seluf(acc);
    }
    float o = fc2b[0];
#pragma unroll
    for (int j = 0; j < HID / 2; ++j) o += z[j] * fc2W[j];
    out[g] = 1.0f / (1.0f + __expf(-o));
}

// ---------------- launch ----------------

extern "C" void kernel_launch(void* const* d_in, const int* in_sizes, int n_in,
                              void* d_out, int out_size, void* d_ws, size_t ws_size,
                              hipStream_t stream) {
    const float* x      = (const float*)d_in[0];
    const float* W1     = (const float*)d_in[1];
    const float* al1    = (const float*)d_in[2];
    const float* ar1    = (const float*)d_in[3];
    const float* b1     = (const float*)d_in[4];
    const float* W2     = (const float*)d_in[5];
    const float* al2    = (const float*)d_in[6];
    const float* ar2    = (const float*)d_in[7];
    const float* b2     = (const float*)d_in[8];
    const float* gateW  = (const float*)d_in[9];
    const float* gateB  = (const float*)d_in[10];
    const float* fc1W   = (const float*)d_in[11];
    const float* fc1b   = (const float*)d_in[12];
    const float* fc2W   = (const float*)d_in[13];
    const float* fc2b   = (const float*)d_in[14];
    const int*   src    = (const int*)d_in[15];
    const int*   dst    = (const int*)d_in[16];
    const int*   ng     = (const int*)d_in[17];
    float* out = (float*)d_out;

    const int N = N_NODES, E = N_EDGES, G = N_GRAPHS;

    // workspace layout
    float* ws = (float*)d_ws;
    float* hA    = ws;                    // N*64 : linear output of current layer
    float* hB    = hA + (size_t)N * HID;  // N*64 : aggregation / activated features
    float* el    = hB + (size_t)N * HID;  // N
    float* er    = el + N;                // N
    float* mbuf  = er + N;                // N
    float* sbuf  = mbuf + N;              // N
    float* exb   = sbuf + N;              // E
    float* gateb = exb + E;               // N
    float* gm    = gateb + N;             // G
    float* gs    = gm + G;                // G
    float* pooled= gs + G;                // G*64

    const int T = 256;
    const float NEG_INF = -INFINITY;

    int gemmBlocks   = ((N / 16) * 4 * 32 + T - 1) / T;     // one wave per 16x16 tile
    int waveNodeBlk  = (N * 32 + T - 1) / T;                // wave per node
    int edgeBlocks   = (E + T - 1) / T;
    int edgeCBlocks  = ((size_t)E * HID + T - 1) / T;
    int nodeCBlocks  = (N * HID + T - 1) / T;
    int nodeBlocks   = (N + T - 1) / T;

    // ===== Layer 1 =====
    gemm_wmma_f32<IN_CH><<<gemmBlocks, T, 0, stream>>>(x, W1, hA, N);
    node_dots<<<waveNodeBlk, T, 0, stream>>>(hA, al1, ar1, el, er, N);
    fill_f32<<<nodeBlocks, T, 0, stream>>>(mbuf, NEG_INF, N);
    fill_f32<<<nodeBlocks, T, 0, stream>>>(sbuf, 0.0f, N);
    fill_f32<<<nodeCBlocks, T, 0, stream>>>(hB, 0.0f, N * HID);
    edge_max<<<edgeBlocks, T, 0, stream>>>(src, dst, el, er, mbuf, E);
    edge_exp<<<edgeBlocks, T, 0, stream>>>(src, dst, el, er, mbuf, exb, sbuf, E);
    edge_agg<<<edgeCBlocks, T, 0, stream>>>(src, dst, exb, sbuf, hA, hB, E);
    bias_selu<<<nodeCBlocks, T, 0, stream>>>(hB, b1, N * HID);

    // ===== Layer 2 =====
    gemm_wmma_f32<HID><<<gemmBlocks, T, 0, stream>>>(hB, W2, hA, N);
    node_dots<<<waveNodeBlk, T, 0, stream>>>(hA, al2, ar2, el, er, N);
    fill_f32<<<nodeBlocks, T, 0, stream>>>(mbuf, NEG_INF, N);
    fill_f32<<<nodeBlocks, T, 0, stream>>>(sbuf, 0.0f, N);
    fill_f32<<<nodeCBlocks, T, 0, stream>>>(hB, 0.0f, N * HID);
    edge_max<<<edgeBlocks, T, 0, stream>>>(src, dst, el, er, mbuf, E);
    edge_exp<<<edgeBlocks, T, 0, stream>>>(src, dst, el, er, mbuf, exb, sbuf, E);
    edge_agg<<<edgeCBlocks, T, 0, stream>>>(src, dst, exb, sbuf, hA, hB, E);
    bias_selu<<<nodeCBlocks, T, 0, stream>>>(hB, b2, N * HID);

    // ===== Global attention pooling =====
    fill_f32<<<1, T, 0, stream>>>(gm, NEG_INF, G);
    fill_f32<<<1, T, 0, stream>>>(gs, 0.0f, G);
    fill_f32<<<(G * HID + T - 1) / T, T, 0, stream>>>(pooled, 0.0f, G * HID);
    gate_compute<<<waveNodeBlk, T, 0, stream>>>(hB, gateW, gateB, ng, gateb, gm, N);
    gate_exp<<<nodeBlocks, T, 0, stream>>>(gateb, gm, gs, ng, N);
    pool_agg<<<nodeCBlocks, T, 0, stream>>>(gateb, gs, hB, ng, pooled, N);

    // ===== Head =====
    head_kernel<<<1, 64, 0, stream>>>(pooled, fc1W, fc1b, fc2W, fc2b, out);
}